// Seq2SeqLSTM_111669150203
// MI455X (gfx1250) — compile-verified
//
#include <hip/hip_runtime.h>
#include <hip/hip_bf16.h>
#include <math.h>

typedef float v2f __attribute__((ext_vector_type(2)));
typedef float v8f __attribute__((ext_vector_type(8)));

#define HD 256      // hidden
#define BB 16       // batch
#define SS 128      // src len
#define TT 128      // tgt len
#define TM1 127     // T-1
#define VOUT 32000
#define G4 1024     // 4*H

__device__ __forceinline__ v8f wmma4(v2f a, v2f b, v8f c) {
    return __builtin_amdgcn_wmma_f32_16x16x4_f32(false, a, false, b, (short)0, c, false, false);
}

__device__ __forceinline__ float sigm(float x) { return 1.0f / (1.0f + __expf(-x)); }

// Generic shared pointer -> 32-bit LDS byte offset (low 32 bits of the
// shared-aperture generic address are the LDS offset).
__device__ __forceinline__ uint32_t lds_off(const void* p) {
    return (uint32_t)(uintptr_t)p;
}

// CDNA5 async global->LDS copy (16B per lane), tracked by ASYNCcnt.
__device__ __forceinline__ void async_g2l_b128(uint32_t ldsoff, const void* g) {
    asm volatile("global_load_async_to_lds_b128 %0, %1, off"
                 :: "v"(ldsoff), "v"(g) : "memory");
}
__device__ __forceinline__ void wait_async0() {
    asm volatile("s_wait_asynccnt 0x0" ::: "memory");
}

// -------------------------------------------------------------------------
// Embedding gathers
// -------------------------------------------------------------------------
__global__ void embed_enc_kernel(const int* __restrict__ src, const float* __restrict__ emb,
                                 float* __restrict__ out) {
    int idx = blockIdx.x * blockDim.x + threadIdx.x;   // BB*SS*HD
    if (idx >= BB * SS * HD) return;
    int h = idx & (HD - 1);
    int bs = idx >> 8;
    out[idx] = emb[(size_t)src[bs] * HD + h];
}

__global__ void embed_dec_kernel(const int* __restrict__ tgt, const float* __restrict__ emb,
                                 float* __restrict__ out) {
    int idx = blockIdx.x * blockDim.x + threadIdx.x;   // BB*TM1*HD
    if (idx >= BB * TM1 * HD) return;
    int h = idx & (HD - 1);
    int m = idx >> 8;                // m = b*127 + t
    int b = m / TM1, t = m % TM1;
    out[idx] = emb[(size_t)tgt[b * TT + t] * HD + h];
}

// -------------------------------------------------------------------------
// Weight transpose: W (4H, H) row-major -> WT (H, 4H) so that WMMA
// B-fragment loads become coalesced dword loads (lanes span n).
// -------------------------------------------------------------------------
__global__ void transpose_w_kernel(const float* __restrict__ W, float* __restrict__ WT) {
    int idx = blockIdx.x * blockDim.x + threadIdx.x;   // G4*HD
    if (idx >= G4 * HD) return;
    int k = idx >> 10, n = idx & (G4 - 1);             // write coalesced
    WT[idx] = W[(size_t)n * HD + k];
}

// -------------------------------------------------------------------------
// Persistent single-workgroup LSTM layer. 512 threads = 16 waves.
// Per step: g = [x_t, h] @ [Wih,Whh]^T + (bih+bhh) via WMMA f32 16x16x4
// (M = batch = 16 exactly). Wave w owns hidden cols j in [w*16, w*16+16):
// acc[0..3] = i/f/g/o gate tiles at n = j, j+256, j+512, j+768, so the
// gate nonlinearity runs entirely in registers. Weights are pre-transposed
// (K-major) so B-fragment loads are coalesced; x_t staged with async copies.
// -------------------------------------------------------------------------
#define XH_LD 516   // words; 516*4 bytes is 16B-aligned, bank stride 4 (conflict-free)

__global__ __launch_bounds__(512) void lstm_layer_kernel(
    const float* __restrict__ X, int steps,
    const float* __restrict__ WihT,   // (HD, G4) K-major
    const float* __restrict__ WhhT,   // (HD, G4) K-major
    const float* __restrict__ bih, const float* __restrict__ bhh,
    const float* __restrict__ h0, const float* __restrict__ c0,
    float* __restrict__ ys, float* __restrict__ hT, float* __restrict__ cT)
{
    __shared__ __align__(16) float xh[BB * XH_LD];   // [m][0:256]=x_t  [m][256:512]=h
    __shared__ __align__(16) float cbuf[BB * HD];
    __shared__ __align__(16) float bias[G4];

    const int tid = threadIdx.x;
    const int lane = tid & 31;
    const int wv = tid >> 5;              // 0..15
    const int mrow = lane & 15;
    const int kh = (lane >> 4) * 2;
    const int ncol = lane & 15;
    const int jcol = wv * 16 + ncol;      // hidden unit owned by this lane

    for (int i = tid; i < G4; i += 512) bias[i] = bih[i] + bhh[i];
    for (int i = tid; i < BB * HD; i += 512) {
        int m = i >> 8, j = i & (HD - 1);
        xh[m * XH_LD + HD + j] = h0 ? h0[m * HD + j] : 0.0f;
        cbuf[m * HD + j]       = c0 ? c0[m * HD + j] : 0.0f;
    }
    __syncthreads();

    for (int t = 0; t < steps; ++t) {
        // async-stage x_t into LDS (X layout: [b][steps][H]); 16B per lane
        for (int i = tid; i < BB * (HD / 4); i += 512) {
            int m = i >> 6, j4 = (i & 63) * 4;
            async_g2l_b128(lds_off(&xh[m * XH_LD + j4]),
                           X + ((size_t)m * steps + t) * HD + j4);
        }
        wait_async0();
        __syncthreads();

        v8f acc[4];
        #pragma unroll
        for (int nt = 0; nt < 4; ++nt) {
            float bv = bias[nt * HD + jcol];
            #pragma unroll
            for (int v = 0; v < 8; ++v) acc[nt][v] = bv;
        }
        // K = 0..255 : x_t @ Wih^T   (B-frags: coalesced dword loads from WihT)
        for (int k0 = 0; k0 < HD; k0 += 4) {
            v2f a = *(const v2f*)&xh[mrow * XH_LD + k0 + kh];
            const float* w0 = WihT + (size_t)(k0 + kh) * G4;
            #pragma unroll
            for (int nt = 0; nt < 4; ++nt) {
                int n = nt * HD + jcol;
                v2f b; b.x = w0[n]; b.y = w0[G4 + n];
                acc[nt] = wmma4(a, b, acc[nt]);
            }
        }
        // K = 256..511 : h @ Whh^T
        for (int k0 = 0; k0 < HD; k0 += 4) {
            v2f a = *(const v2f*)&xh[mrow * XH_LD + HD + k0 + kh];
            const float* w0 = WhhT + (size_t)(k0 + kh) * G4;
            #pragma unroll
            for (int nt = 0; nt < 4; ++nt) {
                int n = nt * HD + jcol;
                v2f b; b.x = w0[n]; b.y = w0[G4 + n];
                acc[nt] = wmma4(a, b, acc[nt]);
            }
        }
        __syncthreads();   // all waves done reading xh before h is overwritten

        // gate nonlinearity in registers; write new h/c
        #pragma unroll
        for (int v = 0; v < 8; ++v) {
            int m = v + ((lane >> 4) << 3);
            float gi = acc[0][v], gf = acc[1][v], gg = acc[2][v], go = acc[3][v];
            float cc = sigm(gf) * cbuf[m * HD + jcol] + sigm(gi) * tanhf(gg);
            float hh = sigm(go) * tanhf(cc);
            cbuf[m * HD + jcol] = cc;
            xh[m * XH_LD + HD + jcol] = hh;
            ys[((size_t)m * steps + t) * HD + jcol] = hh;
        }
        __syncthreads();   // h/c committed before next step's GEMM
    }

    if (hT) {
        for (int i = tid; i < BB * HD; i += 512) {
            int m = i >> 8, j = i & (HD - 1);
            hT[m * HD + j] = xh[m * XH_LD + HD + j];
            cT[m * HD + j] = cbuf[m * HD + j];
        }
    }
}

// -------------------------------------------------------------------------
// Generic WMMA f32 GEMM: C[row_map(m), n] = sum_k A[m,k] * W[n, k] + bias[n]
// A: M x K row-major. W: N rows, stride ldw. Block tile 64(M) x 128(N),
// 8 waves = 1 N-tile x 4 M-subtiles each. Both A (64xKC) and W (128xKC)
// tiles are staged into LDS with async B128 copies (coalesced); fragment
// reads come from LDS with a stride-68 padding (bank-conflict-free v2f).
// -------------------------------------------------------------------------
#define GK_C 64
#define GLD 68

__global__ __launch_bounds__(256) void gemm_wmma_kernel(
    const float* __restrict__ A, const float* __restrict__ W, int ldw,
    const float* __restrict__ bias, float* __restrict__ C, int ldc,
    const int* __restrict__ rmap, int M, int N, int K)
{
    __shared__ __align__(16) float As[64 * GLD];
    __shared__ __align__(16) float Ws[128 * GLD];
    const int tid = threadIdx.x;
    const int lane = tid & 31;
    const int wv = tid >> 5;                  // 0..7
    const int m0 = blockIdx.y * 64;
    const int n0blk = blockIdx.x * 128;
    const int n = n0blk + wv * 16 + (lane & 15);
    const int kh = (lane >> 4) * 2;
    const int mr = lane & 15;

    v8f acc[4];
    float bv = bias ? bias[n] : 0.0f;
    #pragma unroll
    for (int mt = 0; mt < 4; ++mt)
        #pragma unroll
        for (int v = 0; v < 8; ++v) acc[mt][v] = bv;

    for (int kc = 0; kc < K; kc += GK_C) {
        __syncthreads();
        // stage A tile: 64 rows x 64 k (rows >= M left as-is; never stored)
        for (int i = tid; i < 64 * (GK_C / 4); i += 256) {
            int r = i >> 4, k4 = (i & 15) * 4;
            int gm = m0 + r;
            if (gm < M)
                async_g2l_b128(lds_off(&As[r * GLD + k4]),
                               A + (size_t)gm * K + kc + k4);
        }
        // stage W tile: 128 rows x 64 k
        for (int i = tid; i < 128 * (GK_C / 4); i += 256) {
            int r = i >> 4, k4 = (i & 15) * 4;
            async_g2l_b128(lds_off(&Ws[r * GLD + k4]),
                           W + (size_t)(n0blk + r) * ldw + kc + k4);
        }
        wait_async0();
        __syncthreads();
        for (int k0 = 0; k0 < GK_C; k0 += 4) {
            v2f bf = *(const v2f*)&Ws[(wv * 16 + (lane & 15)) * GLD + k0 + kh];
            #pragma unroll
            for (int mt = 0; mt < 4; ++mt) {
                v2f af = *(const v2f*)&As[(mt * 16 + mr) * GLD + k0 + kh];
                acc[mt] = wmma4(af, bf, acc[mt]);
            }
        }
    }

    const int mbase = (lane >> 4) << 3;
    #pragma unroll
    for (int mt = 0; mt < 4; ++mt)
        #pragma unroll
        for (int v = 0; v < 8; ++v) {
            int m = m0 + mt * 16 + v + mbase;
            if (m < M) {
                if (rmap) {
                    // large streamed output: bypass-ish stores, keep L2 for W
                    __builtin_nontemporal_store(acc[mt][v],
                        &C[(size_t)rmap[m] * ldc + n]);
                } else {
                    C[(size_t)m * ldc + n] = acc[mt][v];
                }
            }
        }
}

// -------------------------------------------------------------------------
// Fused attention: per (b,t) block of 256 threads.
// scores[s] = sum_h v_w[h]*tanh(dproj[b,t,h] + eproj[b,s,h])  (attn_b folded
// into dproj); softmax over s; ctx[b,t,h] = sum_s aw[s]*enc_out[b,s,h].
// -------------------------------------------------------------------------
__global__ __launch_bounds__(256) void attn_kernel(
    const float* __restrict__ dproj, const float* __restrict__ eproj,
    const float* __restrict__ enc_out, const float* __restrict__ v_w,
    float* __restrict__ ctx)
{
    __shared__ float dpb[HD], vw[HD], sc[SS], ex[SS];
    const int bt = blockIdx.x;                 // b*127 + t
    const int b = bt / TM1;
    const int tid = threadIdx.x;
    const int lane = tid & 31;
    const int wv = tid >> 5;                   // 0..7

    dpb[tid] = dproj[(size_t)bt * HD + tid];
    vw[tid]  = v_w[tid];
    __syncthreads();

    // Phase A: each wave does 16 s values; lanes span h
    for (int si = 0; si < 16; ++si) {
        int s = wv * 16 + si;
        const float* ep = eproj + ((size_t)b * SS + s) * HD;
        float p = 0.0f;
        #pragma unroll
        for (int j = 0; j < 8; ++j) {
            int h = lane + j * 32;
            p += vw[h] * tanhf(dpb[h] + ep[h]);
        }
        #pragma unroll
        for (int off = 16; off > 0; off >>= 1) p += __shfl_down(p, off, 32);
        if (lane == 0) sc[s] = p;
    }
    __syncthreads();

    // Phase B: softmax (redundant per-thread scans over 128 values in LDS)
    float mx = -INFINITY;
    for (int s = 0; s < SS; ++s) mx = fmaxf(mx, sc[s]);
    if (tid < SS) ex[tid] = expf(sc[tid] - mx);
    __syncthreads();
    float den = 0.0f;
    for (int s = 0; s < SS; ++s) den += ex[s];
    const float inv = 1.0f / den;

    // Phase C: thread owns h = tid
    float a = 0.0f;
    for (int s = 0; s < SS; ++s)
        a = fmaf(ex[s] * inv, enc_out[((size_t)b * SS + s) * HD + tid], a);
    ctx[(size_t)bt * HD + tid] = a;
}

// -------------------------------------------------------------------------
// Small helpers
// -------------------------------------------------------------------------
__global__ void pack_kernel(const float* __restrict__ dec_out, const float* __restrict__ ctx,
                            float* __restrict__ Ap) {
    int idx = blockIdx.x * blockDim.x + threadIdx.x;   // (B*TM1) * 512
    if (idx >= BB * TM1 * 2 * HD) return;
    int k = idx & 511, m = idx >> 9;
    Ap[idx] = (k < HD) ? dec_out[(size_t)m * HD + k] : ctx[(size_t)m * HD + (k - HD)];
}

__global__ void rmap_kernel(int* __restrict__ rmap) {
    int m = blockIdx.x * blockDim.x + threadIdx.x;
    if (m >= BB * TM1) return;
    rmap[m] = (m / TM1) * TT + (m % TM1) + 1;
}

__global__ void zero_row0_kernel(float* __restrict__ out) {
    int idx = blockIdx.x * blockDim.x + threadIdx.x;   // BB * VOUT
    if (idx >= BB * VOUT) return;
    int b = idx / VOUT, j = idx % VOUT;
    out[(size_t)b * TT * VOUT + j] = 0.0f;
}

// -------------------------------------------------------------------------
// Launcher
// -------------------------------------------------------------------------
extern "C" void kernel_launch(void* const* d_in, const int* in_sizes, int n_in,
                              void* d_out, int out_size, void* d_ws, size_t ws_size,
                              hipStream_t stream) {
    (void)in_sizes; (void)n_in; (void)out_size; (void)ws_size;

    const int*   src     = (const int*)d_in[0];
    const int*   tgt     = (const int*)d_in[1];
    const float* enc_emb = (const float*)d_in[2];
    const float* dec_emb = (const float*)d_in[3];
    const float* eW[2][4] = {{(const float*)d_in[4],(const float*)d_in[5],(const float*)d_in[6],(const float*)d_in[7]},
                             {(const float*)d_in[8],(const float*)d_in[9],(const float*)d_in[10],(const float*)d_in[11]}};
    const float* dW[2][4] = {{(const float*)d_in[12],(const float*)d_in[13],(const float*)d_in[14],(const float*)d_in[15]},
                             {(const float*)d_in[16],(const float*)d_in[17],(const float*)d_in[18],(const float*)d_in[19]}};
    const float* attn_W = (const float*)d_in[20];
    const float* attn_b = (const float*)d_in[21];
    const float* v_w    = (const float*)d_in[22];
    const float* out_W  = (const float*)d_in[23];
    const float* out_b  = (const float*)d_in[24];
    float* out = (float*)d_out;

    // workspace carving
    size_t off = 0;
    auto fbuf = [&](size_t nf) { float* p = (float*)((char*)d_ws + off);
                                 off = (off + nf * sizeof(float) + 255) & ~(size_t)255; return p; };
    float* enc_x   = fbuf((size_t)BB * SS * HD);
    float* dec_x   = fbuf((size_t)BB * TM1 * HD);
    float* ys0     = fbuf((size_t)BB * SS * HD);      // enc L0 out, reused for dec L0 out
    float* enc_out = fbuf((size_t)BB * SS * HD);
    float* dec_out = fbuf((size_t)BB * TM1 * HD);
    float* dproj   = fbuf((size_t)BB * TM1 * HD);
    float* eproj   = fbuf((size_t)BB * SS * HD);
    float* ctxb    = fbuf((size_t)BB * TM1 * HD);
    float* Apack   = fbuf((size_t)BB * TM1 * 2 * HD);
    float* wt_ih   = fbuf((size_t)HD * G4);           // transposed weights (reused per pass)
    float* wt_hh   = fbuf((size_t)HD * G4);
    float* encH0   = fbuf(BB * HD);
    float* encC0   = fbuf(BB * HD);
    float* encH1   = fbuf(BB * HD);
    float* encC1   = fbuf(BB * HD);
    int*   rmap    = (int*)((char*)d_ws + off); off += ((size_t)BB * TM1 * sizeof(int) + 255) & ~(size_t)255;

    const int TW_GRID = (G4 * HD + 255) / 256;

    // 1) embeddings
    embed_enc_kernel<<<(BB*SS*HD + 255)/256, 256, 0, stream>>>(src, enc_emb, enc_x);
    embed_dec_kernel<<<(BB*TM1*HD + 255)/256, 256, 0, stream>>>(tgt, dec_emb, dec_x);

    // 2) encoder LSTM (2 layers), capture finals for decoder init
    transpose_w_kernel<<<TW_GRID, 256, 0, stream>>>(eW[0][0], wt_ih);
    transpose_w_kernel<<<TW_GRID, 256, 0, stream>>>(eW[0][1], wt_hh);
    lstm_layer_kernel<<<1, 512, 0, stream>>>(enc_x, SS, wt_ih, wt_hh, eW[0][2], eW[0][3],
                                             nullptr, nullptr, ys0, encH0, encC0);
    transpose_w_kernel<<<TW_GRID, 256, 0, stream>>>(eW[1][0], wt_ih);
    transpose_w_kernel<<<TW_GRID, 256, 0, stream>>>(eW[1][1], wt_hh);
    lstm_layer_kernel<<<1, 512, 0, stream>>>(ys0,  SS, wt_ih, wt_hh, eW[1][2], eW[1][3],
                                             nullptr, nullptr, enc_out, encH1, encC1);

    // 3) decoder LSTM (2 layers), init from encoder finals
    transpose_w_kernel<<<TW_GRID, 256, 0, stream>>>(dW[0][0], wt_ih);
    transpose_w_kernel<<<TW_GRID, 256, 0, stream>>>(dW[0][1], wt_hh);
    lstm_layer_kernel<<<1, 512, 0, stream>>>(dec_x, TM1, wt_ih, wt_hh, dW[0][2], dW[0][3],
                                             encH0, encC0, ys0, nullptr, nullptr);
    transpose_w_kernel<<<TW_GRID, 256, 0, stream>>>(dW[1][0], wt_ih);
    transpose_w_kernel<<<TW_GRID, 256, 0, stream>>>(dW[1][1], wt_hh);
    lstm_layer_kernel<<<1, 512, 0, stream>>>(ys0,  TM1, wt_ih, wt_hh, dW[1][2], dW[1][3],
                                             encH1, encC1, dec_out, nullptr, nullptr);

    // 4) attention projections (attn_b folded into dproj)
    gemm_wmma_kernel<<<dim3(HD/128, (BB*TM1 + 63)/64), 256, 0, stream>>>(
        dec_out, attn_W, 2*HD, attn_b, dproj, HD, nullptr, BB*TM1, HD, HD);
    gemm_wmma_kernel<<<dim3(HD/128, (BB*SS + 63)/64), 256, 0, stream>>>(
        enc_out, attn_W + HD, 2*HD, nullptr, eproj, HD, nullptr, BB*SS, HD, HD);

    // 5) fused scores/softmax/context
    attn_kernel<<<BB*TM1, 256, 0, stream>>>(dproj, eproj, enc_out, v_w, ctxb);

    // 6) output projection: logits = [dec_out, ctx] @ out_W^T + out_b
    pack_kernel<<<(BB*TM1*2*HD + 255)/256, 256, 0, stream>>>(dec_out, ctxb, Apack);
    rmap_kernel<<<(BB*TM1 + 255)/256, 256, 0, stream>>>(rmap);
    zero_row0_kernel<<<(BB*VOUT + 255)/256, 256, 0, stream>>>(out);
    gemm_wmma_kernel<<<dim3(VOUT/128, (BB*TM1 + 63)/64), 256, 0, stream>>>(
        Apack, out_W, 2*HD, out_b, out, VOUT, rmap, BB*TM1, VOUT, 2*HD);
}